// PatchMatch_55387898250141
// MI455X (gfx1250) — compile-verified
//
#include <hip/hip_runtime.h>
#include <cstdint>

#define SDIM 256
#define CCH  96
#define NBATCH 2
#define NPIX (NBATCH * SDIM * SDIM)
// per-pixel LDS stride for center features: 104 halves = 208 B (13x16B) ->
// ds_load_b128 lane stride of 52 banks => at worst 2-way conflict.
#define FSTRIDE 104

// ---------------- packed f16 helpers (wave32 v_pk_* path) ----------------
typedef _Float16 h2 __attribute__((ext_vector_type(2)));

__device__ __forceinline__ h2 u2h(uint32_t u) { union { uint32_t u; h2 h; } c; c.u = u; return c.h; }
__device__ __forceinline__ uint32_t h2u(h2 h) { union { uint32_t u; h2 h; } c; c.h = h; return c.u; }
__device__ __forceinline__ h2 habs(h2 d) { return u2h(h2u(d) & 0x7FFF7FFFu); }

__device__ __forceinline__ void async_b128_to_lds(uint32_t lds_off, const void* gaddr) {
  unsigned long long ga = (unsigned long long)(uintptr_t)gaddr;
  asm volatile("global_load_async_to_lds_b128 %0, %1, off" :: "v"(lds_off), "v"(ga) : "memory");
}
__device__ __forceinline__ void async_b32_to_lds(uint32_t lds_off, const void* gaddr) {
  unsigned long long ga = (unsigned long long)(uintptr_t)gaddr;
  asm volatile("global_load_async_to_lds_b32 %0, %1, off" :: "v"(lds_off), "v"(ga) : "memory");
}
__device__ __forceinline__ void wait_async0() {
  asm volatile("s_wait_asynccnt 0x0" ::: "memory");
}

// ---------------- Kernel 1: NCHW f32 -> NHWC f16 repack ----------------
// One-time 50MB stream; NT loads so it does not evict the L2-resident featT.
__global__ void __launch_bounds__(256) k_pack_nhwc(const float* __restrict__ feat,
                                                   _Float16* __restrict__ featT) {
  __shared__ _Float16 tile[CCH * 64];  // 12 KB
  const int tid = threadIdx.x;
  const int p0  = blockIdx.x * 64;          // flat pixel base (stays within one batch)
  const int b   = p0 / (SDIM * SDIM);
  const int hw0 = p0 % (SDIM * SDIM);
  const int cq = tid / 64, px = tid % 64;
  #pragma unroll
  for (int cc = 0; cc < 24; ++cc) {
    int c = cq * 24 + cc;
    float v = __builtin_nontemporal_load(
        feat + ((size_t)(b * CCH + c)) * (SDIM * SDIM) + hw0 + px);
    tile[c * 64 + px] = (_Float16)v;
  }
  __syncthreads();
  _Float16* out = featT + (size_t)p0 * CCH;
  #pragma unroll
  for (int k = 0; k < 24; ++k) {
    int flat = k * 256 + tid;          // 0 .. 6143
    int opx = flat / CCH, oc = flat % CCH;
    out[flat] = tile[oc * 64 + opx];
  }
}

// ---------------- candidate strength: bilinear gather + 3x3 block-SAD ----------------
// Center features come from LDS (myf, staged via async-to-LDS); sampled vector a[48]
// is built register-resident from 4x 192B b128 gathers (all L2-resident after repack).
__device__ __forceinline__ float cand_strength(const _Float16* __restrict__ featT, int b,
                                               float lx, float ly, float cox, float coy,
                                               const uint4* __restrict__ myf) {
  const float smax = (float)(SDIM - 1);
  float rx = fminf(fmaxf(lx + cox, 0.f), smax);
  float ry = fminf(fmaxf(ly + coy, 0.f), smax);
  float xf = floorf(rx), yf = floorf(ry);
  float wx = rx - xf, wy = ry - yf;
  int x0 = (int)xf, y0 = (int)yf;
  int x1 = (x0 + 1 < SDIM) ? x0 + 1 : SDIM - 1;
  int y1 = (y0 + 1 < SDIM) ? y0 + 1 : SDIM - 1;
  _Float16 h00 = (_Float16)((1.f - wx) * (1.f - wy));
  _Float16 h01 = (_Float16)(wx * (1.f - wy));
  _Float16 h10 = (_Float16)((1.f - wx) * wy);
  _Float16 h11 = (_Float16)(wx * wy);
  h2 W00 = {h00, h00}, W01 = {h01, h01}, W10 = {h10, h10}, W11 = {h11, h11};
  const _Float16* base = featT + (size_t)b * (SDIM * SDIM * CCH);
  const uint4* p00 = (const uint4*)(base + ((size_t)y0 * SDIM + x0) * CCH);
  const uint4* p01 = (const uint4*)(base + ((size_t)y0 * SDIM + x1) * CCH);
  const uint4* p10 = (const uint4*)(base + ((size_t)y1 * SDIM + x0) * CCH);
  const uint4* p11 = (const uint4*)(base + ((size_t)y1 * SDIM + x1) * CCH);

  h2 a[48];  // full 96-channel bilinear sample, packed f16x2
  #pragma unroll
  for (int q = 0; q < 12; ++q) {
    uint4 t = p00[q];
    a[q*4+0] = W00 * u2h(t.x); a[q*4+1] = W00 * u2h(t.y);
    a[q*4+2] = W00 * u2h(t.z); a[q*4+3] = W00 * u2h(t.w);
  }
  #pragma unroll
  for (int q = 0; q < 12; ++q) {
    uint4 t = p01[q];
    a[q*4+0] += W01 * u2h(t.x); a[q*4+1] += W01 * u2h(t.y);
    a[q*4+2] += W01 * u2h(t.z); a[q*4+3] += W01 * u2h(t.w);
  }
  #pragma unroll
  for (int q = 0; q < 12; ++q) {
    uint4 t = p10[q];
    a[q*4+0] += W10 * u2h(t.x); a[q*4+1] += W10 * u2h(t.y);
    a[q*4+2] += W10 * u2h(t.z); a[q*4+3] += W10 * u2h(t.w);
  }
  #pragma unroll
  for (int q = 0; q < 12; ++q) {
    uint4 t = p11[q];
    a[q*4+0] += W11 * u2h(t.x); a[q*4+1] += W11 * u2h(t.y);
    a[q*4+2] += W11 * u2h(t.z); a[q*4+3] += W11 * u2h(t.w);
  }

  float best = 3.4e38f;
  #pragma unroll
  for (int i = 0; i < 3; ++i) {        // feat-side 32-channel block (from LDS)
    uint32_t fr[16];
    #pragma unroll
    for (int k = 0; k < 4; ++k) {      // ds_load_b128
      uint4 t = myf[i * 4 + k];
      fr[k*4+0] = t.x; fr[k*4+1] = t.y; fr[k*4+2] = t.z; fr[k*4+3] = t.w;
    }
    #pragma unroll
    for (int j = 0; j < 3; ++j) {      // sample-side 32-channel block
      h2 acc = {(_Float16)0.f, (_Float16)0.f};
      #pragma unroll
      for (int k = 0; k < 16; ++k) acc += habs(a[j * 16 + k] - u2h(fr[k]));
      float s = (float)acc.x + (float)acc.y;
      best = fminf(best, s);
    }
  }
  return -best * (1.f / 32.f);  // max of block means == -min SAD / 32
}

// branchless online-softmax update (scale x1000); r == 1.0 when max unchanged
__device__ __forceinline__ void smax_update(float s, float ccx, float ccy,
                                            float& m, float& wsum, float& sx, float& sy) {
  float newm = fmaxf(m, s);
  float r = __expf((m - newm) * 1000.f);
  float w = __expf((s - newm) * 1000.f);
  wsum = wsum * r + w;
  sx = sx * r + ccx * w;
  sy = sy * r + ccy * w;
  m = newm;
}

__device__ __forceinline__ void finish_fuse(float& nx, float& ny, float sx, float sy,
                                            float wsum, float lx, float ly) {
  nx = sx / wsum; ny = sy / wsum;
  ny = fminf(fmaxf(ny + ly, 0.f), (float)(SDIM - 1)) - ly;
  nx = fminf(fmaxf(nx + lx, 0.f), (float)(SDIM - 1)) - lx;
}

// ---------------- Kernel 2/4: propagate (13 cands) ----------------
__global__ void __launch_bounds__(256) k_eval_prop(const _Float16* __restrict__ featT,
                                                   const float* __restrict__ ox_in,
                                                   const float* __restrict__ oy_in,
                                                   float* __restrict__ ox_out,
                                                   float* __restrict__ oy_out) {
  __shared__ alignas(16) _Float16 shf[256 * FSTRIDE];  // 52 KB center-feature tile
  __shared__ float shx[20 * 20];                       // offset halo tiles (roll wraps)
  __shared__ float shy[20 * 20];
  __shared__ float scx[13 * 256];                      // candidate stash (lane-major)
  __shared__ float scy[13 * 256];
  const int b = blockIdx.z;
  const int bx = blockIdx.x * 16, by = blockIdx.y * 16;
  const int tx = threadIdx.x, ty = threadIdx.y;
  const int tid = ty * 16 + tx;
  const int x = bx + tx, y = by + ty;
  const size_t pid = (size_t)(b * SDIM + y) * SDIM + x;

  // --- async stage: this pixel's 192B feature vector (12x b128) ---
  const _Float16* src = featT + pid * CCH;
  uint32_t lbase = (uint32_t)(uintptr_t)(shf + (size_t)tid * FSTRIDE);
  #pragma unroll
  for (int q = 0; q < 12; ++q) async_b128_to_lds(lbase + q * 16, src + q * 8);

  // --- async stage: 20x20 offset halo (jnp.roll wraps mod 256) ---
  const float* oxp = ox_in + (size_t)b * SDIM * SDIM;
  const float* oyp = oy_in + (size_t)b * SDIM * SDIM;
  #pragma unroll
  for (int pass = 0; pass < 2; ++pass) {
    int idx = pass * 256 + tid;
    if (idx < 400) {
      int tly = idx / 20, tlx = idx % 20;
      int gy = (by - 2 + tly + SDIM) & (SDIM - 1);
      int gx = (bx - 2 + tlx + SDIM) & (SDIM - 1);
      async_b32_to_lds((uint32_t)(uintptr_t)(&shx[idx]), oxp + gy * SDIM + gx);
      async_b32_to_lds((uint32_t)(uintptr_t)(&shy[idx]), oyp + gy * SDIM + gx);
    }
  }
  wait_async0();
  __syncthreads();

  // --- build 13 propagation candidates, stash in LDS (conflict-free lane-major) ---
  auto T = [&](int dy, int dx) { return (ty + 2 + dy) * 20 + (tx + 2 + dx); };
  #define ST(k, vx, vy) scx[(k)*256 + tid] = (vx); scy[(k)*256 + tid] = (vy);
  #define C1(k, dy, dx) ST(k, shx[T(dy,dx)], shy[T(dy,dx)])
  #define C2(k, dy, dx, dy2, dx2) \
      ST(k, 2.f * shx[T(dy,dx)] - shx[T(dy2,dx2)], 2.f * shy[T(dy,dx)] - shy[T(dy2,dx2)])
  C1(0, 0, 0)
  C1(1, 0, -1)  C1(2, 0, 1)  C1(3, -1, 0)  C1(4, 1, 0)
  C2(5, 0, -1, 0, -2)   C2(6, 0, 1, 0, 2)
  C2(7, -1, 0, -2, 0)   C2(8, 1, 0, 2, 0)
  C2(9, -1, -1, -2, -2) C2(10, 1, 1, 2, 2)
  C2(11, -1, 1, -2, 2)  C2(12, 1, -1, 2, -2)
  #undef C1
  #undef C2
  #undef ST

  // --- rolled candidate loop + branchless online softmax ---
  const float lx = (float)x, ly = (float)y;
  const uint4* myf = (const uint4*)(shf + (size_t)tid * FSTRIDE);
  float m = -3.4e38f, wsum = 0.f, sx = 0.f, sy = 0.f;
  #pragma unroll 1
  for (int k = 0; k < 13; ++k) {
    float ccx = scx[k * 256 + tid], ccy = scy[k * 256 + tid];
    float s = cand_strength(featT, b, lx, ly, ccx, ccy, myf);
    smax_update(s, ccx, ccy, m, wsum, sx, sy);
  }
  float nx, ny;
  finish_fuse(nx, ny, sx, sy, wsum, lx, ly);
  ox_out[pid] = nx;
  oy_out[pid] = ny;
}

// ---------------- Kernel 3/5: random sample (5 cands) ----------------
__global__ void __launch_bounds__(256) k_eval_rand(const _Float16* __restrict__ featT,
                                                   const float* __restrict__ ox_in,
                                                   const float* __restrict__ oy_in,
                                                   const float* __restrict__ noise_it,
                                                   float window,
                                                   float* __restrict__ ox_out,
                                                   float* __restrict__ oy_out) {
  __shared__ alignas(16) _Float16 shf[256 * FSTRIDE];
  const int b = blockIdx.z;
  const int tx = threadIdx.x, ty = threadIdx.y;
  const int tid = ty * 16 + tx;
  const int x = blockIdx.x * 16 + tx;
  const int y = blockIdx.y * 16 + ty;
  const size_t pid = (size_t)(b * SDIM + y) * SDIM + x;

  const _Float16* src = featT + pid * CCH;
  uint32_t lbase = (uint32_t)(uintptr_t)(shf + (size_t)tid * FSTRIDE);
  #pragma unroll
  for (int q = 0; q < 12; ++q) async_b128_to_lds(lbase + q * 16, src + q * 8);

  const float lx = (float)x, ly = (float)y;
  const float smax = (float)(SDIM - 1);
  float ox = ox_in[pid], oy = oy_in[pid];
  float n0 = noise_it[((size_t)b * 2 + 0) * SDIM * SDIM + (size_t)y * SDIM + x];
  float n1 = noise_it[((size_t)b * 2 + 1) * SDIM * SDIM + (size_t)y * SDIM + x];
  float minx = fmaxf(ox - window, -lx), maxx = fminf(ox + window, smax - lx);
  float miny = fmaxf(oy - window, -ly), maxy = fminf(oy + window, smax - ly);
  float xr0 = minx + (maxx - minx) * 0.5f * n0;
  float xr1 = minx + (maxx - minx) * 0.5f * (1.f + n1);
  float yr0 = miny + (maxy - miny) * 0.5f * n0;
  float yr1 = miny + (maxy - miny) * 0.5f * (1.f + n1);
  float cx[5] = {xr0, xr1, xr0, xr1, ox};
  float cy[5] = {yr0, yr0, yr1, yr1, oy};

  wait_async0();  // only this lane's staged bytes are read -> no barrier needed
  const uint4* myf = (const uint4*)(shf + (size_t)tid * FSTRIDE);

  float m = -3.4e38f, wsum = 0.f, sx = 0.f, sy = 0.f;
  #pragma unroll
  for (int k = 0; k < 5; ++k) {
    float s = cand_strength(featT, b, lx, ly, cx[k], cy[k], myf);
    smax_update(s, cx[k], cy[k], m, wsum, sx, sy);
  }
  float nx, ny;
  finish_fuse(nx, ny, sx, sy, wsum, lx, ly);
  ox_out[pid] = nx;
  oy_out[pid] = ny;
}

// ---------------- host-side launcher ----------------
extern "C" void kernel_launch(void* const* d_in, const int* in_sizes, int n_in,
                              void* d_out, int out_size, void* d_ws, size_t ws_size,
                              hipStream_t stream) {
  const float* feat  = (const float*)d_in[0];  // [2,96,256,256]
  const float* off_x = (const float*)d_in[1];  // [2,1,256,256]
  const float* off_y = (const float*)d_in[2];
  const float* noise = (const float*)d_in[3];  // [2,2,2,256,256]

  char* ws = (char*)d_ws;
  _Float16* featT = (_Float16*)ws;             // 24 MB NHWC f16 (L2-resident)
  const size_t featBytes = (size_t)NPIX * CCH * sizeof(_Float16);
  float* oxA = (float*)(ws + featBytes);
  float* oyA = oxA + NPIX;
  float* oxB = oyA + NPIX;
  float* oyB = oxB + NPIX;

  float* out_x = (float*)d_out;                // tuple return order: (nox, noy)
  float* out_y = out_x + NPIX;

  const size_t noiseStride = (size_t)NBATCH * 2 * SDIM * SDIM;

  k_pack_nhwc<<<NPIX / 64, 256, 0, stream>>>(feat, featT);

  dim3 blk(16, 16);
  dim3 grd(SDIM / 16, SDIM / 16, NBATCH);
  // iter 0 (window 128)
  k_eval_prop<<<grd, blk, 0, stream>>>(featT, off_x, off_y, oxA, oyA);
  k_eval_rand<<<grd, blk, 0, stream>>>(featT, oxA, oyA, noise + 0 * noiseStride, 128.f, oxB, oyB);
  // iter 1 (window 32)
  k_eval_prop<<<grd, blk, 0, stream>>>(featT, oxB, oyB, oxA, oyA);
  k_eval_rand<<<grd, blk, 0, stream>>>(featT, oxA, oyA, noise + 1 * noiseStride, 32.f, out_x, out_y);
}